// SSMKernel_71906342470088
// MI455X (gfx1250) — compile-verified
//
#include <hip/hip_runtime.h>

// S4D-style diagonal SSM scan, chunked into 16x16x16 fp32 WMMA matmuls.
// BATCH=8, SEQ=512, D_MODEL=2048, D_STATE=16.
//
// Per channel d (time-invariant dt):
//   dA[n] = exp(dt*exp(A_log)), dB[n] = dt*B[d,n]
//   h_t = dA*h_{t-1} + dB*x_t ; y_t = C . h_t
// Chunk T=16:
//   Yin[s,j] = sum_n C dB dA^{s-j}  (s>=j)    [Toeplitz m[p]]
//   W[n,j]   = dB[n] dA[n]^{15-j}
//   U[s,n]   = C[n] dA[n]^{s+1}
//   P = W @ X ; Hprev = scan(dA^16, P) ; Y = Yin @ X + U @ Hprev

#define SEQ     512
#define DMODEL  2048
#define NST     16
#define CHUNK   16
#define NCHUNK  (SEQ / CHUNK)   // 32
#define DT_CH   8               // channels per block
#define XA_C    196             // LDS chunk stride (floats): 4c mod 64 -> conflict-free
#define XA_J    12              // LDS j stride (floats): 8ch + 4 pad, rows 16B aligned
#define YSTR    516             // Y staging stride
#define PSTRIDE 33              // 32 + pad

typedef float v2f __attribute__((ext_vector_type(2)));
typedef float v8f __attribute__((ext_vector_type(8)));

#if defined(__has_builtin)
#if __has_builtin(__builtin_amdgcn_global_load_async_to_lds_b128) && \
    __has_builtin(__builtin_amdgcn_s_wait_asynccnt)
#define USE_ASYNC_LDS 1
#endif
#endif

#if defined(USE_ASYNC_LDS)
// Builtin prototype (from hipcc diagnostic): param0 = vector_size(16) int
// in AS1 ("__device__"), param1 = same pointee in AS3, then imm offset, imm cpol.
typedef int v4i_vs __attribute__((vector_size(16)));
typedef __attribute__((address_space(1))) v4i_vs g_v4i;
typedef __attribute__((address_space(3))) v4i_vs l_v4i;
// AMDGPU generic LDS address = {aperture_hi32, lds_offset_lo32}; truncating to
// 32 bits recovers the AS3 offset. Integer->pointer casts avoid addrspacecast
// legality questions.
__device__ __forceinline__ l_v4i* lds_cast(const void* p) {
  return (l_v4i*)(unsigned int)(unsigned long long)p;
}
__device__ __forceinline__ g_v4i* gbl_cast(const void* p) {
  return (g_v4i*)(unsigned long long)p;
}
#endif

__global__ __launch_bounds__(128) void ssm_chunk_wmma(
    const float* __restrict__ x,      // [B, SEQ, D]
    const float* __restrict__ A_log,  // [D, N]
    const float* __restrict__ Bmat,   // [D, N]
    const float* __restrict__ Cmat,   // [D, N]
    const float* __restrict__ log_dt, // [D]
    float* __restrict__ y)            // [B, SEQ, D]
{
  __shared__ float sXa[NCHUNK * XA_C];         // [c][j][ch]; reused as Y stage
  __shared__ float sPw[DT_CH * 17 * 16];       // [ch][p][n]  dA^p, p=0..16
  __shared__ float sM[DT_CH * 16];             // [ch][p]     Toeplitz m[p]
  __shared__ float sDB[DT_CH * 16];            // [ch][n]
  __shared__ float sC[DT_CH * 16];             // [ch][n]
  __shared__ float sG[DT_CH * 16];             // [ch][n]     dA^16
  __shared__ float sP[DT_CH * 16 * PSTRIDE];   // [ch][n][c]  P, then Hprev

  const int tid = threadIdx.x;
  const int b   = blockIdx.y;
  const int d0  = blockIdx.x * DT_CH;
  const float* xb = x + (size_t)b * SEQ * DMODEL + d0;

#if defined(USE_ASYNC_LDS)
  // ---- Issue async X staging first: overlaps with exp-heavy Phase A.
  // 16B granule = 4 channels at one t; tile = 512 t x 2 granules.
#pragma unroll
  for (int k = 0; k < (DT_CH * SEQ / 4) / 128; ++k) {
    const int i  = tid + k * 128;
    const int v4 = (i & 1) * 4;
    const int t  = i >> 1;
    const int c  = t >> 4, j = t & 15;
    __builtin_amdgcn_global_load_async_to_lds_b128(
        gbl_cast(xb + (size_t)t * DMODEL + v4),
        lds_cast(&sXa[c * XA_C + j * XA_J + v4]), 0, 0);
  }
#endif

  // ---- Phase A: per-channel params + power table (128 threads = 8ch x 16n)
  {
    const int ch = tid >> 4, n = tid & 15;
    const int d  = d0 + ch;
    const float dtv = expf(log_dt[d]);               // DT_SCALE == 1
    const float av  = expf(A_log[d * NST + n]);
    const float da  = expf(dtv * av);
    sDB[ch * 16 + n] = dtv * Bmat[d * NST + n];
    sC[ch * 16 + n]  = Cmat[d * NST + n];
    float pw = 1.0f;
#pragma unroll
    for (int p = 0; p <= 16; ++p) {
      sPw[(ch * 17 + p) * 16 + n] = pw;
      if (p == 16) sG[ch * 16 + n] = pw;
      pw *= da;
    }
  }

#if defined(USE_ASYNC_LDS)
  __builtin_amdgcn_s_wait_asynccnt(0);
#else
  // ---- Fallback: synchronous X staging (float4 loads, scatter to LDS)
  for (int i = tid; i < DT_CH * SEQ / 4; i += 128) {
    const int v4 = (i & 1) * 4;
    const int t  = i >> 1;
    const int c  = t >> 4, j = t & 15;
    const float4 val = *reinterpret_cast<const float4*>(xb + (size_t)t * DMODEL + v4);
    const int base = c * XA_C + j * XA_J + v4;
    sXa[base + 0] = val.x;
    sXa[base + 1] = val.y;
    sXa[base + 2] = val.z;
    sXa[base + 3] = val.w;
  }
#endif
  __syncthreads();

  // ---- Toeplitz vector m[p] = sum_n C*dB*dA^p (128 threads = 8ch x 16p)
  {
    const int ch = tid >> 4, p = tid & 15;
    float s = 0.0f;
#pragma unroll
    for (int n = 0; n < 16; ++n)
      s += sC[ch * 16 + n] * sDB[ch * 16 + n] * sPw[(ch * 17 + p) * 16 + n];
    sM[ch * 16 + p] = s;
  }
  // (sM only read after the post-B1 barrier; no barrier needed here)

  const int lane  = tid & 31;
  const int wv    = tid >> 5;
  const int row   = lane & 15;             // A-row / B-col / D-col
  const int hi    = (lane >> 4) << 1;      // per-lane K offset (ISA layout)
  const int rbase = (lane < 16) ? 0 : 8;   // D-matrix row base

  // ---- B1: P[n,c] = W @ X. Compile-time trip counts -> no exec masking.
#pragma unroll
  for (int cc = 0; cc < 2; ++cc) {
    const int ch = wv + 4 * cc;
    const float dbn = sDB[ch * 16 + row];
    v2f aW[4];
#pragma unroll
    for (int kk = 0; kk < 4; ++kk) {
      const int j0 = 4 * kk + hi;
      aW[kk][0] = dbn * sPw[(ch * 17 + (15 - j0)) * 16 + row];
      aW[kk][1] = dbn * sPw[(ch * 17 + (14 - j0)) * 16 + row];
    }
#pragma unroll
    for (int ct = 0; ct < 2; ++ct) {
      const int c = ct * 16 + row;
      v8f acc = {0.f, 0.f, 0.f, 0.f, 0.f, 0.f, 0.f, 0.f};
#pragma unroll
      for (int kk = 0; kk < 4; ++kk) {
        const int j0 = 4 * kk + hi;
        v2f bb;
        bb[0] = sXa[c * XA_C + j0 * XA_J + ch];
        bb[1] = sXa[c * XA_C + (j0 + 1) * XA_J + ch];
        acc = __builtin_amdgcn_wmma_f32_16x16x4_f32(false, aW[kk], false, bb,
                                                    (short)0, acc, false, false);
      }
#pragma unroll
      for (int v = 0; v < 8; ++v)
        sP[ch * (16 * PSTRIDE) + (rbase + v) * PSTRIDE + c] = acc[v];
    }
  }
  __syncthreads();

  // ---- B2: inter-chunk scan in place: P[c] -> Hprev[c] (state before chunk c)
  {
    const int ch = tid >> 4, n = tid & 15;
    const float g = sG[ch * 16 + n];
    float h = 0.0f;
    float* prow = &sP[ch * (16 * PSTRIDE) + n * PSTRIDE];
#pragma unroll 4
    for (int c = 0; c < NCHUNK; ++c) {
      const float pv = prow[c];
      prow[c] = h;
      h = fmaf(g, h, pv);
    }
  }
  __syncthreads();

  // ---- B3: Y = Yin @ X + U @ Hprev (accumulators held across barrier)
  v8f accs[2][2];
#pragma unroll
  for (int cc = 0; cc < 2; ++cc) {
    const int ch = wv + 4 * cc;
    v2f aY[4], aU[4];
#pragma unroll
    for (int kk = 0; kk < 4; ++kk) {
      const int j0 = 4 * kk + hi;
      aY[kk][0] = (row >= j0)     ? sM[ch * 16 + row - j0]     : 0.0f;
      aY[kk][1] = (row >= j0 + 1) ? sM[ch * 16 + row - j0 - 1] : 0.0f;
      aU[kk][0] = sC[ch * 16 + j0]     * sPw[(ch * 17 + row + 1) * 16 + j0];
      aU[kk][1] = sC[ch * 16 + j0 + 1] * sPw[(ch * 17 + row + 1) * 16 + j0 + 1];
    }
#pragma unroll
    for (int ct = 0; ct < 2; ++ct) {
      const int c = ct * 16 + row;
      v8f acc = {0.f, 0.f, 0.f, 0.f, 0.f, 0.f, 0.f, 0.f};
#pragma unroll
      for (int kk = 0; kk < 4; ++kk) {     // intra-chunk: causal Toeplitz
        const int j0 = 4 * kk + hi;
        v2f bb;
        bb[0] = sXa[c * XA_C + j0 * XA_J + ch];
        bb[1] = sXa[c * XA_C + (j0 + 1) * XA_J + ch];
        acc = __builtin_amdgcn_wmma_f32_16x16x4_f32(false, aY[kk], false, bb,
                                                    (short)0, acc, false, false);
      }
#pragma unroll
      for (int kk = 0; kk < 4; ++kk) {     // carry-in: U @ Hprev
        const int n0 = 4 * kk + hi;
        v2f bb;
        bb[0] = sP[ch * (16 * PSTRIDE) + n0 * PSTRIDE + c];
        bb[1] = sP[ch * (16 * PSTRIDE) + (n0 + 1) * PSTRIDE + c];
        acc = __builtin_amdgcn_wmma_f32_16x16x4_f32(false, aU[kk], false, bb,
                                                    (short)0, acc, false, false);
      }
      accs[cc][ct] = acc;
    }
  }
  __syncthreads();   // all sXa reads done -> safe to reuse as Y stage

  // ---- Stage Y in LDS ([ch][t]) for coalesced global stores
  float* sY = sXa;
#pragma unroll
  for (int cc = 0; cc < 2; ++cc) {
    const int ch = wv + 4 * cc;
#pragma unroll
    for (int ct = 0; ct < 2; ++ct) {
      const int c = ct * 16 + row;
#pragma unroll
      for (int v = 0; v < 8; ++v)
        sY[ch * YSTR + c * CHUNK + rbase + v] = accs[cc][ct][v];
    }
  }
  __syncthreads();

  float* yb = y + (size_t)b * SEQ * DMODEL + d0;
  for (int i = tid; i < DT_CH * SEQ / 4; i += 128) {
    const int v4 = (i & 1) * 4;
    const int t  = i >> 1;
    float4 val;
    val.x = sY[(v4 + 0) * YSTR + t];
    val.y = sY[(v4 + 1) * YSTR + t];
    val.z = sY[(v4 + 2) * YSTR + t];
    val.w = sY[(v4 + 3) * YSTR + t];
    *reinterpret_cast<float4*>(yb + (size_t)t * DMODEL + v4) = val;
  }
}

extern "C" void kernel_launch(void* const* d_in, const int* in_sizes, int n_in,
                              void* d_out, int out_size, void* d_ws, size_t ws_size,
                              hipStream_t stream) {
  (void)in_sizes; (void)n_in; (void)out_size; (void)d_ws; (void)ws_size;
  const float* x      = (const float*)d_in[0];
  const float* A_log  = (const float*)d_in[1];
  const float* Bmat   = (const float*)d_in[2];
  const float* Cmat   = (const float*)d_in[3];
  const float* log_dt = (const float*)d_in[4];
  float* yo = (float*)d_out;

  dim3 grid(DMODEL / DT_CH, 8 /*BATCH*/);
  ssm_chunk_wmma<<<grid, 128, 0, stream>>>(x, A_log, Bmat, Cmat, log_dt, yo);
}